// GIN_25692494364721
// MI455X (gfx1250) — compile-verified
//
#include <hip/hip_runtime.h>

#define H 64
#define TSTR 68  // padded LDS row stride (floats): 64 data + 4 pad, 16B-aligned rows, conflict-free

typedef __attribute__((ext_vector_type(2))) float v2f;
typedef __attribute__((ext_vector_type(8))) float v8f;
typedef __attribute__((ext_vector_type(4))) unsigned int u32x4;
typedef __attribute__((ext_vector_type(8))) int i32x8;
typedef __attribute__((ext_vector_type(4))) int i32x4;

#if __has_builtin(__builtin_amdgcn_tensor_load_to_lds)
#define HAVE_TDM 1
#else
#define HAVE_TDM 0
#endif

__device__ __forceinline__ v8f wmma_f32(v2f a, v2f b, v8f c) {
  // D = A(16x4 f32) * B(4x16 f32) + C(16x16 f32)
  return __builtin_amdgcn_wmma_f32_16x16x4_f32(false, a, false, b, (short)0, c, false, false);
}

#if HAVE_TDM
// TDM: async DMA of a 2D f32 tile (width x rows, row stride `stride` elems) from global
// memory into LDS at lds_off, inserting 4 pad DWORDs after every 64 DWORDs so the tile
// lands directly in the padded TSTR=68 layout. Tracked by TENSORcnt (per-wave).
__device__ __forceinline__ void tdm_load_tile_padded(const float* gsrc, unsigned lds_off,
                                                     unsigned width, unsigned rows,
                                                     unsigned stride) {
  unsigned long long ga = (unsigned long long)(const void*)gsrc;
  u32x4 g0;
  g0[0] = 1u;                                                  // count=1 (valid), user mode
  g0[1] = lds_off;                                             // D#.lds_addr
  g0[2] = (unsigned)(ga & 0xffffffffu);                        // global_addr[31:0]
  g0[3] = (unsigned)((ga >> 32) & 0x01ffffffu) | 0x80000000u;  // global_addr[56:32] | type=2
  i32x8 g1;
  // data_size=2 (4B), pad_enable=1, pad_interval=5 (64 DWORDs), pad_amount=3 (4 DWORDs)
  g1[0] = (2 << 16) | (1 << 20) | (5 << 22) | (3 << 25);
  g1[1] = (int)((width & 0xffffu) << 16);                      // tensor_dim0[15:0]
  g1[2] = (int)((width >> 16) & 0xffffu) | (int)((rows & 0xffffu) << 16);  // dim0 hi | dim1 lo
  g1[3] = (int)((rows >> 16) & 0xffffu) | (int)((width & 0xffffu) << 16);  // dim1 hi | tile_dim0
  g1[4] = (int)(rows & 0xffffu);                               // tile_dim1 (tile_dim2=0)
  g1[5] = (int)stride;                                         // tensor_dim0_stride[31:0]
  g1[6] = 0;
  g1[7] = 0;
  i32x4 z4 = {0, 0, 0, 0};
  i32x8 z8 = {0, 0, 0, 0, 0, 0, 0, 0};
  // 6-arg variant (clang-23 / therock-10.0 headers): groups 2,3 zero, extra group, cpol=0
  __builtin_amdgcn_tensor_load_to_lds(g0, g1, z4, z4, z8, 0);
}
#endif

__global__ void zero_kernel(float* __restrict__ p, int n) {
  int i = blockIdx.x * blockDim.x + threadIdx.x;
  if (i < n) p[i] = 0.0f;
}

// agg[dst[e]] += x[src[e]]  -- one thread per (edge, 4-feature chunk)
__global__ void scatter_kernel(const float* __restrict__ x, const int* __restrict__ src,
                               const int* __restrict__ dst, float* __restrict__ agg, int E) {
  int tid = blockIdx.x * blockDim.x + threadIdx.x;
  int e = tid >> 4;
  if (e >= E) return;
  int c = (tid & 15) << 2;
  int s = src[e];
  int d = dst[e];
  const float4 v = *(const float4*)(x + (size_t)s * H + c);
  float* o = agg + (size_t)d * H + c;
  atomicAdd(o + 0, v.x);
  atomicAdd(o + 1, v.y);
  atomicAdd(o + 2, v.z);
  atomicAdd(o + 3, v.w);
}

// Fused GIN MLP: h = ReLU((x+agg)@W1 + b1)@W2 + b2, plus per-column sum/sumsq for BN.
// One wave computes a 16-row tile; 8 waves (8 tiles) per 256-thread block.
// Weights live in LDS pair-interleaved ((W[k][n],W[k+1][n]) adjacent -> 1x ds_load_b64 per
// WMMA B-operand) with a k-parity column swizzle so both half-waves hit disjoint banks.
__global__ void __launch_bounds__(256)
gin_mlp_kernel(const float* __restrict__ x, const float* __restrict__ agg,
               const float* __restrict__ W1, const float* __restrict__ b1,
               const float* __restrict__ W2, const float* __restrict__ b2,
               float* __restrict__ hout, float* __restrict__ ssum, float* __restrict__ ssq,
               int N, int ntiles) {
  __shared__ float sW1p[H * H];
  __shared__ float sW2p[H * H];
  __shared__ float sB[2 * H];
  __shared__ float sX[8][16 * TSTR];   // staged x tile, per wave (padded layout)
  __shared__ float sAG[8][16 * TSTR];  // staged agg tile, per wave (padded layout)
  __shared__ float sT[8][16 * TSTR];   // ReLU intermediate, per wave

  const int t = threadIdx.x;
  const int wave = t >> 5;
  const int lane = t & 31;
  const int tile = blockIdx.x * 8 + wave;
  const bool active = (tile < ntiles);  // wave-uniform
  const int rowbase = tile * 16;
  const float* X = &sX[wave][0];
  const float* AG = &sAG[wave][0];
  float* T = &sT[wave][0];

#if HAVE_TDM
  {
    // Clamp so tail waves issue a harmless duplicate load (TDM ignores EXEC; issue from
    // every wave unconditionally with uniform operands).
    int rb = rowbase;
    int rbmax = (ntiles - 1) * 16;
    if (rb > rbmax) rb = rbmax;
    rb = __builtin_amdgcn_readfirstlane(rb);
    int rows = N - rb; if (rows > 16) rows = 16;
    rows = __builtin_amdgcn_readfirstlane(rows);
    unsigned ldsX = __builtin_amdgcn_readfirstlane((unsigned)(unsigned long long)(const void*)X);
    unsigned ldsA = __builtin_amdgcn_readfirstlane((unsigned)(unsigned long long)(const void*)AG);
    tdm_load_tile_padded(x + (size_t)rb * H, ldsX, H, (unsigned)rows, H);
    tdm_load_tile_padded(agg + (size_t)rb * H, ldsA, H, (unsigned)rows, H);
  }
#endif

  // Cooperative weight stage + pair-interleave + bank swizzle.
  // Element W[k][n] -> s[(k>>1)*128 + ((n + ((k>>1)&1)*16) & 63)*2 + (k&1)]
  for (int i = t; i < H * H; i += 256) {
    int k = i >> 6;
    int n = i & 63;
    int p = k >> 1;
    int dst = (p << 7) + ((((n + ((p & 1) << 4))) & 63) << 1) + (k & 1);
    sW1p[dst] = W1[i];
    sW2p[dst] = W2[i];
  }
  if (t < H) sB[t] = b1[t];
  else if (t < 2 * H) sB[t] = b2[t - H];

#if !HAVE_TDM
  if (active) {
    for (int i = lane; i < 256; i += 32) {
      int r = i >> 4;
      int c = (i & 15) << 2;
      int row = rowbase + r;
      float4 xv = {0.f, 0.f, 0.f, 0.f}, av = {0.f, 0.f, 0.f, 0.f};
      if (row < N) {
        size_t off = (size_t)row * H + c;
        xv = *(const float4*)(x + off);
        av = *(const float4*)(agg + off);
      }
      *(float4*)(&sX[wave][0] + r * TSTR + c) = xv;
      *(float4*)(&sAG[wave][0] + r * TSTR + c) = av;
    }
  }
#endif

  __syncthreads();
#if HAVE_TDM
  __builtin_amdgcn_s_wait_tensorcnt(0);
#endif

  const int m  = lane & 15;          // A row / B,C,D column within 16-block
  const int kb = (lane >> 4) << 1;   // K sub-offset: 0 (lanes 0-15) or 2 (lanes 16-31)
  const int hi = (lane >> 4) << 4;   // column swizzle for this half-wave's k-pair parity
  const int cs0 = (((0 << 4) + m + hi) & 63) << 1;  // swizzled column*2 per n-block
  const int cs1 = (((1 << 4) + m + hi) & 63) << 1;
  const int cs2 = (((2 << 4) + m + hi) & 63) << 1;
  const int cs3 = (((3 << 4) + m + hi) & 63) << 1;

  v8f c0 = {}, c1 = {}, c2 = {}, c3 = {};
  if (active) {
    // GEMM1: (x+agg) @ W1
#pragma unroll
    for (int k = 0; k < 16; ++k) {
      const int kk = (k << 2) + kb;
      const int pb = ((kk >> 1) << 7);
      v2f a = *(const v2f*)(X + m * TSTR + kk) + *(const v2f*)(AG + m * TSTR + kk);
      c0 = wmma_f32(a, *(const v2f*)(sW1p + pb + cs0), c0);
      c1 = wmma_f32(a, *(const v2f*)(sW1p + pb + cs1), c1);
      c2 = wmma_f32(a, *(const v2f*)(sW1p + pb + cs2), c2);
      c3 = wmma_f32(a, *(const v2f*)(sW1p + pb + cs3), c3);
    }
    // bias + ReLU -> T (A-operand layout for GEMM2)
#pragma unroll
    for (int r = 0; r < 8; ++r) {
      int row = ((lane >> 4) << 3) + r;
      T[row * TSTR + m]      = fmaxf(c0[r] + sB[m],      0.f);
      T[row * TSTR + 16 + m] = fmaxf(c1[r] + sB[16 + m], 0.f);
      T[row * TSTR + 32 + m] = fmaxf(c2[r] + sB[32 + m], 0.f);
      T[row * TSTR + 48 + m] = fmaxf(c3[r] + sB[48 + m], 0.f);
    }
  }
  __syncthreads();

  v8f d0 = {}, d1 = {}, d2 = {}, d3 = {};
  if (active) {
    // GEMM2: T @ W2
#pragma unroll
    for (int k = 0; k < 16; ++k) {
      const int kk = (k << 2) + kb;
      const int pb = ((kk >> 1) << 7);
      v2f a = *(const v2f*)(T + m * TSTR + kk);
      d0 = wmma_f32(a, *(const v2f*)(sW2p + pb + cs0), d0);
      d1 = wmma_f32(a, *(const v2f*)(sW2p + pb + cs1), d1);
      d2 = wmma_f32(a, *(const v2f*)(sW2p + pb + cs2), d2);
      d3 = wmma_f32(a, *(const v2f*)(sW2p + pb + cs3), d3);
    }
    // bias, store pre-BN h, accumulate column sum/sumsq
    float ps0 = 0.f, pq0 = 0.f, ps1 = 0.f, pq1 = 0.f;
    float ps2 = 0.f, pq2 = 0.f, ps3 = 0.f, pq3 = 0.f;
#pragma unroll
    for (int r = 0; r < 8; ++r) {
      int row = rowbase + ((lane >> 4) << 3) + r;
      if (row < N) {
        size_t off = (size_t)row * H;
        float v0 = d0[r] + sB[H + m];
        float v1 = d1[r] + sB[H + 16 + m];
        float v2 = d2[r] + sB[H + 32 + m];
        float v3 = d3[r] + sB[H + 48 + m];
        hout[off + m]      = v0; ps0 += v0; pq0 += v0 * v0;
        hout[off + 16 + m] = v1; ps1 += v1; pq1 += v1 * v1;
        hout[off + 32 + m] = v2; ps2 += v2; pq2 += v2 * v2;
        hout[off + 48 + m] = v3; ps3 += v3; pq3 += v3 * v3;
      }
    }
    atomicAdd(&ssum[m], ps0);      atomicAdd(&ssq[m], pq0);
    atomicAdd(&ssum[16 + m], ps1); atomicAdd(&ssq[16 + m], pq1);
    atomicAdd(&ssum[32 + m], ps2); atomicAdd(&ssq[32 + m], pq2);
    atomicAdd(&ssum[48 + m], ps3); atomicAdd(&ssq[48 + m], pq3);
  }
}

// Fold BN stats into fused scale/shift: y = h*scale + shift
__global__ void bn_stats_kernel(const float* __restrict__ ssum, const float* __restrict__ ssq,
                                const float* __restrict__ g, const float* __restrict__ be,
                                float* __restrict__ scale, float* __restrict__ shift, float Nf) {
  int c = threadIdx.x;
  if (c >= H) return;
  float inv = 1.0f / Nf;
  float mu = ssum[c] * inv;
  float var = ssq[c] * inv - mu * mu;
  float sc = rsqrtf(var + 1e-5f) * g[c];
  scale[c] = sc;
  shift[c] = be[c] - mu * sc;
}

__global__ void bn_apply_kernel(const float* __restrict__ h, const float* __restrict__ scale,
                                const float* __restrict__ shift, float* __restrict__ y, int n16) {
  int tid = blockIdx.x * blockDim.x + threadIdx.x;
  if (tid >= n16) return;
  int c = (tid & 15) << 2;
  size_t off = ((size_t)(tid >> 4)) * H + c;
  float4 v = *(const float4*)(h + off);
  float4 o;
  o.x = fmaxf(v.x * scale[c + 0] + shift[c + 0], 0.f);
  o.y = fmaxf(v.y * scale[c + 1] + shift[c + 1], 0.f);
  o.z = fmaxf(v.z * scale[c + 2] + shift[c + 2], 0.f);
  o.w = fmaxf(v.w * scale[c + 3] + shift[c + 3], 0.f);
  *(float4*)(y + off) = o;
}

__global__ void pool_kernel(const float* __restrict__ y, const int* __restrict__ batch,
                            float* __restrict__ gsum, float* __restrict__ gcnt, int N) {
  int tid = blockIdx.x * blockDim.x + threadIdx.x;
  int n = tid >> 4;
  if (n >= N) return;
  int c = (tid & 15) << 2;
  int b = batch[n];
  float4 v = *(const float4*)(y + (size_t)n * H + c);
  float* o = gsum + b * H + c;
  atomicAdd(o + 0, v.x);
  atomicAdd(o + 1, v.y);
  atomicAdd(o + 2, v.z);
  atomicAdd(o + 3, v.w);
  if ((tid & 15) == 0) atomicAdd(&gcnt[b], 1.0f);
}

__global__ void pool_fin_kernel(const float* __restrict__ gsum, const float* __restrict__ gcnt,
                                float* __restrict__ out, int GH) {
  int i = blockIdx.x * blockDim.x + threadIdx.x;
  if (i >= GH) return;
  out[i] = gsum[i] / fmaxf(gcnt[i >> 6], 1.0f);
}

extern "C" void kernel_launch(void* const* d_in, const int* in_sizes, int n_in,
                              void* d_out, int out_size, void* d_ws, size_t ws_size,
                              hipStream_t stream) {
  const float* x     = (const float*)d_in[0];
  const int*   ei    = (const int*)d_in[1];
  const int*   batch = (const int*)d_in[2];
  const float* W1_0 = (const float*)d_in[3];
  const float* b1_0 = (const float*)d_in[4];
  const float* W2_0 = (const float*)d_in[5];
  const float* b2_0 = (const float*)d_in[6];
  const float* g_0  = (const float*)d_in[7];
  const float* be_0 = (const float*)d_in[8];
  const float* W1_1 = (const float*)d_in[9];
  const float* b1_1 = (const float*)d_in[10];
  const float* W2_1 = (const float*)d_in[11];
  const float* b2_1 = (const float*)d_in[12];
  const float* g_1  = (const float*)d_in[13];
  const float* be_1 = (const float*)d_in[14];

  const int N  = in_sizes[0] / H;
  const int E  = in_sizes[1] / 2;
  const int G  = out_size / H - N;
  const int NH = N * H;
  const int ntiles = (N + 15) / 16;

  float* buf0  = (float*)d_ws;       // agg, then aliased as pre-BN h (disjoint row ownership)
  float* ybuf  = buf0 + NH;          // layer-1 output
  float* ssum  = ybuf + NH;
  float* ssq   = ssum + H;
  float* scale = ssq + H;
  float* shift = scale + H;
  float* gsum  = shift + H;
  float* gcnt  = gsum + G * H;       // contiguous after gsum

  float* outx = (float*)d_out;       // final node features [N,H]
  float* outg = outx + NH;           // graph means [G,H]

  const int* src = ei;
  const int* dst = ei + E;

  dim3 blk(256);
  auto layer = [&](const float* xin, const float* W1, const float* b1, const float* W2,
                   const float* b2, const float* g, const float* be, float* yout) {
    zero_kernel<<<(NH + 255) / 256, blk, 0, stream>>>(buf0, NH);
    zero_kernel<<<1, blk, 0, stream>>>(ssum, 2 * H);
    scatter_kernel<<<(E * 16 + 255) / 256, blk, 0, stream>>>(xin, src, dst, buf0, E);
    gin_mlp_kernel<<<(ntiles + 7) / 8, blk, 0, stream>>>(xin, buf0, W1, b1, W2, b2,
                                                         buf0, ssum, ssq, N, ntiles);
    bn_stats_kernel<<<1, 64, 0, stream>>>(ssum, ssq, g, be, scale, shift, (float)N);
    bn_apply_kernel<<<(N * 16 + 255) / 256, blk, 0, stream>>>(buf0, scale, shift, yout, N * 16);
  };

  layer(x,    W1_0, b1_0, W2_0, b2_0, g_0, be_0, ybuf);
  layer(ybuf, W1_1, b1_1, W2_1, b2_1, g_1, be_1, outx);

  zero_kernel<<<1, blk, 0, stream>>>(gsum, G * H + G);
  pool_kernel<<<(N * 16 + 255) / 256, blk, 0, stream>>>(outx, batch, gsum, gcnt, N);
  pool_fin_kernel<<<(G * H + 255) / 256, blk, 0, stream>>>(gsum, gcnt, outg, G * H);
}